// Totalloss6_72928544686596
// MI455X (gfx1250) — compile-verified
//
#include <hip/hip_runtime.h>
#include <math.h>

typedef __attribute__((ext_vector_type(2))) float v2f;
typedef __attribute__((ext_vector_type(8))) float v8f;

#define N_S    16
#define NCLS   20
#define TLEN   1000
#define CDIM   2048
#define KSNIP  64
#define T_NCE  0.1f
#define NEGINF (-1e30f)

// ---------------- Kernel 1: M0 = embeddings.mean(axis=1)  [16,2048] ----------------
// thread per (i, ch4): sums 1000 float4 at stride 2048 floats (coalesced 512B/warp).
__global__ void k_mean_time(const float* __restrict__ emb, float* __restrict__ M0) {
    int idx = blockIdx.x * blockDim.x + threadIdx.x;      // 16*512 threads
    int i  = idx / (CDIM / 4);
    int c4 = idx % (CDIM / 4);
    const float4* p = (const float4*)(emb + (size_t)i * TLEN * CDIM) + c4;
    float4 s = {0.f, 0.f, 0.f, 0.f};
    for (int t = 0; t < TLEN; ++t) {
        float4 v = p[(size_t)t * (CDIM / 4)];
        s.x += v.x; s.y += v.y; s.z += v.z; s.w += v.w;
    }
    const float inv = 1.0f / (float)TLEN;
    float4 o; o.x = s.x * inv; o.y = s.y * inv; o.z = s.z * inv; o.w = s.w * inv;
    ((float4*)M0)[idx] = o;
}

// ---------------- Kernel 2: generic contiguous row mean (len multiple of 4) ----------------
// one block per row; used for T0 (16000 rows of 2048) and Tea (1024 rows of 2048).
__global__ void k_row_mean(const float* __restrict__ x, float* __restrict__ out, int len) {
    __shared__ float sm[256];
    const float4* row = (const float4*)(x + (size_t)blockIdx.x * len);
    float s = 0.f;
    for (int j = threadIdx.x; j < len / 4; j += blockDim.x) {
        float4 v = row[j];
        s += v.x + v.y + v.z + v.w;
    }
    sm[threadIdx.x] = s;
    __syncthreads();
    for (int off = 128; off > 0; off >>= 1) {
        if ((int)threadIdx.x < off) sm[threadIdx.x] += sm[threadIdx.x + off];
        __syncthreads();
    }
    if (threadIdx.x == 0) out[blockIdx.x] = sm[0] / (float)len;
}

// ---------------- Kernel 3: Mcat rows = [mean(EA), mean(HA), mean(EB), mean(HB)] over ksnip ----------------
__global__ void k_mean_snip(const float* __restrict__ ea, const float* __restrict__ ha,
                            const float* __restrict__ eb, const float* __restrict__ hb,
                            float* __restrict__ Mcat) {
    int idx = blockIdx.x * blockDim.x + threadIdx.x;      // 4*16*512 threads
    int arr = idx / (N_S * (CDIM / 4));
    int rem = idx % (N_S * (CDIM / 4));
    int i   = rem / (CDIM / 4);
    int c4  = rem % (CDIM / 4);
    const float* base = (arr == 0) ? ea : (arr == 1) ? ha : (arr == 2) ? eb : hb;
    const float4* p = (const float4*)(base + (size_t)i * KSNIP * CDIM) + c4;
    float4 s = {0.f, 0.f, 0.f, 0.f};
    for (int k = 0; k < KSNIP; ++k) {
        float4 v = p[(size_t)k * (CDIM / 4)];
        s.x += v.x; s.y += v.y; s.z += v.z; s.w += v.w;
    }
    const float inv = 1.0f / (float)KSNIP;
    int row = arr * N_S + i;
    float4 o; o.x = s.x * inv; o.y = s.y * inv; o.z = s.z * inv; o.w = s.w * inv;
    ((float4*)Mcat)[(size_t)row * (CDIM / 4) + c4] = o;
}

// ---------------- Kernel 4: raw gram matrices via V_WMMA_F32_16X16X4_F32 ----------------
// 19 one-wave jobs: job0 G0 (M0, K=2048); job1 GT (T0, K=1000); job2 GTe (Tea, K=64);
// jobs 3..18: 4x4 tiling of 64x64 gram of Mcat (K=2048).
// Layouts (ISA 7.12.2): A 16x4 f32: lane<16 holds A[l,k],A[l,k+1]; lane>=16 holds A[l,k+2],A[l,k+3].
// B 4x16 f32: rows striped across lanes: vgpr0 = {K=k (lanes0-15), K=k+2 (lanes16-31)}, vgpr1 = {k+1, k+3}.
// C/D 16x16 f32: vgpr r = row (r + 8*half), col = lane&15.
__global__ void k_gram_wmma(const float* __restrict__ M0, const float* __restrict__ T0,
                            const float* __restrict__ Tea, const float* __restrict__ Mcat,
                            float* __restrict__ G0, float* __restrict__ GT,
                            float* __restrict__ GTe, float* __restrict__ Gf) {
    int job = blockIdx.x;
    const float* src; float* dst; int ld, K, mb, nb, lddst;
    if (job == 0)      { src = M0;  dst = G0;  ld = CDIM;  K = CDIM;  mb = 0; nb = 0; lddst = 16; }
    else if (job == 1) { src = T0;  dst = GT;  ld = TLEN;  K = TLEN;  mb = 0; nb = 0; lddst = 16; }
    else if (job == 2) { src = Tea; dst = GTe; ld = KSNIP; K = KSNIP; mb = 0; nb = 0; lddst = 16; }
    else {
        int t = job - 3;
        src = Mcat; dst = Gf; ld = CDIM; K = CDIM;
        mb = (t / 4) * 16; nb = (t % 4) * 16; lddst = 64;
    }
    int lane = threadIdx.x;
    int half = lane >> 4;
    int l    = lane & 15;
    const float* arow = src + (size_t)(mb + l) * ld + 2 * half;
    const float* brow = src + (size_t)(nb + l) * ld + 2 * half;
    v8f acc = {};
    for (int k = 0; k < K; k += 4) {
        v2f a; a.x = arow[k]; a.y = arow[k + 1];
        v2f b; b.x = brow[k]; b.y = brow[k + 1];
        acc = __builtin_amdgcn_wmma_f32_16x16x4_f32(
            /*neg_a=*/false, a, /*neg_b=*/false, b,
            /*c_mod=*/(short)0, acc, /*reuse_a=*/false, /*reuse_b=*/false);
    }
#pragma unroll
    for (int r = 0; r < 8; ++r) {
        int M = r + 8 * half;
        dst[(size_t)(mb + M) * lddst + (nb + l)] = acc[r];
    }
}

// ---------------- Kernel 5: final scalar assembly (one block, 512 threads) ----------------
__device__ __forceinline__ float block_reduce(float v, float* red, int tid) {
    red[tid] = v;
    __syncthreads();
    for (int off = 256; off > 0; off >>= 1) {
        if (tid < off) red[tid] += red[tid + off];
        __syncthreads();
    }
    float r = red[0];
    __syncthreads();
    return r;
}

__global__ void k_final(const float* __restrict__ vs, const float* __restrict__ label,
                        const float* __restrict__ G0, const float* __restrict__ GT,
                        const float* __restrict__ GTe, const float* __restrict__ Gf,
                        float* __restrict__ out) {
    __shared__ float lab[N_S * NCLS];
    __shared__ float rowsum[N_S];
    __shared__ float Dins[N_S * N_S];
    __shared__ float Dfeat[N_S * N_S];
    __shared__ float n0s[N_S];
    __shared__ float nfs[64];
    __shared__ float G0n[N_S * N_S];
    __shared__ float Gfn[64 * 64];
    __shared__ float red[512];
    const int tid = threadIdx.x;

    for (int j = tid; j < N_S * NCLS; j += 512) lab[j] = label[j];
    if (tid < N_S) n0s[tid] = fmaxf(sqrtf(G0[tid * 16 + tid]), 1e-12f);
    if (tid < 64)  nfs[tid] = fmaxf(sqrtf(Gf[tid * 64 + tid]), 1e-12f);
    __syncthreads();

    if (tid < N_S) {
        float s = 0.f;
        for (int c = 0; c < NCLS; ++c) s += lab[tid * NCLS + c];
        rowsum[tid] = s;
    }
    if (tid < 256) {
        int i = tid >> 4, j = tid & 15;
        float ni = fmaxf(sqrtf(GT[i * 16 + i]), 1e-12f);
        float nj = fmaxf(sqrtf(GT[j * 16 + j]), 1e-12f);
        Dins[tid] = 1.0f - GT[i * 16 + j] / (ni * nj);
        float mi = fmaxf(sqrtf(GTe[i * 16 + i]), 1e-12f);
        float mj = fmaxf(sqrtf(GTe[j * 16 + j]), 1e-12f);
        Dfeat[tid] = 1.0f - GTe[i * 16 + j] / (mi * mj);
    }
    for (int e = tid; e < 256; e += 512) {
        int a = e >> 4, b = e & 15;
        G0n[e] = G0[e] / (n0s[a] * n0s[b]);
    }
    for (int e = tid; e < 64 * 64; e += 512) {
        int a = e >> 6, b = e & 63;
        Gfn[e] = Gf[e] / (nfs[a] * nfs[b]);
    }
    __syncthreads();

    // ---- BCE (loss_cls) ----
    float bcev = 0.f;
    if (tid < N_S * NCLS) {
        int i = tid / NCLS;
        float y   = lab[tid] / rowsum[i];
        float p   = vs[tid];
        float lp  = fmaxf(logf(p), -100.0f);
        float l1p = fmaxf(logf(1.0f - p), -100.0f);
        bcev = -(y * lp + (1.0f - y) * l1p);
    }

    // ---- separation loss (uses raw norms, no clamp) ----
    float sepv = 0.f;
    if (tid < N_S) {
        float act = fmaxf(150.0f - sqrtf(Gf[tid * 64 + tid]), 0.0f);          // ||mean(EA_i)||
        float bkg = sqrtf(Gf[(32 + tid) * 64 + (32 + tid)]);                   // ||mean(EB_i)||
        float v = act + bkg;
        sepv = v * v;
    }

    // ---- instance contrast: 320 (c,i) pairs ----
    float insv = 0.f;
    if (tid < NCLS * N_S) {
        int c = tid >> 4, i = tid & 15;
        unsigned mm = 0; int cnt = 0;
        for (int j = 0; j < N_S; ++j) {
            bool b = lab[j * NCLS + c] > 0.5f;
            mm |= (unsigned)b << j; cnt += (int)b;
        }
        bool active = ((mm >> i) & 1u) && (cnt > 1);
        float rowmax = Dins[i * 16 + 0];
        for (int j = 1; j < N_S; ++j) rowmax = fmaxf(rowmax, Dins[i * 16 + j]);
        float best = INFINITY; int jb = 0;
        for (int j = 0; j < N_S; ++j) {
            bool ok = ((mm >> j) & 1u) && (j != i);
            float dv = ok ? Dins[i * 16 + j] : INFINITY;
            if (dv < best) { best = dv; jb = j; }               // first-min (argmin)
        }
        int p = (best < rowmax) ? jb : i;
        float l_pos = G0n[p * 16 + i] / T_NCE;
        float mx = l_pos;
        for (int m = 0; m < N_S; ++m) {
            float v = ((mm >> m) & 1u) ? NEGINF : G0n[m * 16 + i] / T_NCE;
            mx = fmaxf(mx, v);
        }
        float s = expf(l_pos - mx);
        for (int m = 0; m < N_S; ++m) {
            float v = ((mm >> m) & 1u) ? NEGINF : G0n[m * 16 + i] / T_NCE;
            s += expf(v - mx);
        }
        insv = active ? (mx + logf(s) - l_pos) : 0.0f;
    }

    // ---- feature contrast: 320 (c,i) pairs on 64x64 Npool gram ----
    float featv = 0.f;
    if (tid < NCLS * N_S) {
        int c = tid >> 4, i = tid & 15;
        unsigned mm = 0; int cnt = 0;
        for (int j = 0; j < N_S; ++j) {
            bool b = lab[j * NCLS + c] > 0.5f;
            mm |= (unsigned)b << j; cnt += (int)b;
        }
        bool active = ((mm >> i) & 1u) && (cnt > 1);
        float rowmax = Dfeat[i * 16 + 0];
        for (int j = 1; j < N_S; ++j) rowmax = fmaxf(rowmax, Dfeat[i * 16 + j]);
        float best = INFINITY; int jb = 0;
        for (int j = 0; j < N_S; ++j) {
            bool ok = ((mm >> j) & 1u) && (j != i);
            float dv = ok ? Dfeat[i * 16 + j] : INFINITY;
            if (dv < best) { best = dv; jb = j; }
        }
        int p1 = (best < rowmax) ? jb : (N_S + i);               // Mea[j] or Mha[i] -> Npool row
        float l_pos1 = Gfn[p1 * 64 + i] / T_NCE;                 // qn = unit(Mea_i) = Npool row i
        float l_pos2 = Gfn[i * 64 + (16 + i)] / T_NCE;           // qn = unit(Mha_i), kn = unit(Mea_i)
        float mx1 = l_pos1, mx2 = l_pos2;
        for (int m = 0; m < 64; ++m) {
            bool isneg = (m < 32) ? !((mm >> (m & 15)) & 1u) : true;
            float v1 = isneg ? Gfn[m * 64 + i] / T_NCE        : NEGINF;
            float v2 = isneg ? Gfn[m * 64 + (16 + i)] / T_NCE : NEGINF;
            mx1 = fmaxf(mx1, v1); mx2 = fmaxf(mx2, v2);
        }
        float s1 = expf(l_pos1 - mx1), s2 = expf(l_pos2 - mx2);
        for (int m = 0; m < 64; ++m) {
            bool isneg = (m < 32) ? !((mm >> (m & 15)) & 1u) : true;
            float v1 = isneg ? Gfn[m * 64 + i] / T_NCE        : NEGINF;
            float v2 = isneg ? Gfn[m * 64 + (16 + i)] / T_NCE : NEGINF;
            s1 += expf(v1 - mx1); s2 += expf(v2 - mx2);
        }
        float loss = (mx1 + logf(s1) - l_pos1) + (mx2 + logf(s2) - l_pos2);
        featv = active ? loss : 0.0f;
    }

    float bce_sum  = block_reduce(bcev,  red, tid);
    float sep_sum  = block_reduce(sepv,  red, tid);
    float ins_sum  = block_reduce(insv,  red, tid);
    float feat_sum = block_reduce(featv, red, tid);

    if (tid == 0) {
        float loss_cls  = bce_sum / (float)(N_S * NCLS);
        float loss_sep  = sep_sum / (float)N_S;
        float loss_ins  = ins_sum / (float)N_S;
        float loss_feat = feat_sum / (float)N_S;
        float total = loss_cls + 0.0f * loss_sep + 0.0f * loss_feat + 0.01f * loss_ins;
        out[0] = total; out[1] = loss_cls; out[2] = loss_sep; out[3] = loss_ins; out[4] = loss_feat;
    }
}

// ---------------- launcher ----------------
extern "C" void kernel_launch(void* const* d_in, const int* in_sizes, int n_in,
                              void* d_out, int out_size, void* d_ws, size_t ws_size,
                              hipStream_t stream) {
    (void)in_sizes; (void)n_in; (void)out_size; (void)ws_size;
    const float* vs  = (const float*)d_in[0];   // [16,20]
    const float* lab = (const float*)d_in[1];   // [16,20]
    const float* emb = (const float*)d_in[2];   // [16,1000,2048]
    const float* EA  = (const float*)d_in[3];   // [16,64,2048]
    const float* EB  = (const float*)d_in[4];
    const float* HA  = (const float*)d_in[5];
    const float* HB  = (const float*)d_in[6];
    float* out = (float*)d_out;

    float* ws   = (float*)d_ws;
    float* M0   = ws;                 // 16*2048      = 32768
    float* T0   = M0   + 32768;       // 16*1000      = 16000 (padded to 16384)
    float* Mcat = T0   + 16384;       // 64*2048      = 131072
    float* Tea  = Mcat + 131072;      // 16*64        = 1024
    float* G0   = Tea  + 1024;        // 16*16
    float* GT   = G0   + 256;         // 16*16
    float* GTe  = GT   + 256;         // 16*16
    float* Gf   = GTe  + 256;         // 64*64

    k_mean_time<<<32,    256, 0, stream>>>(emb, M0);
    k_row_mean <<<16000, 256, 0, stream>>>(emb, T0, CDIM);
    k_mean_snip<<<128,   256, 0, stream>>>(EA, HA, EB, HB, Mcat);   // Npool order: Mea, Mha, Meb, Mhb
    k_row_mean <<<1024,  256, 0, stream>>>(EA, Tea, CDIM);
    k_gram_wmma<<<19,    32,  0, stream>>>(M0, T0, Tea, Mcat, G0, GT, GTe, Gf);
    k_final    <<<1,     512, 0, stream>>>(vs, lab, G0, GT, GTe, Gf, out);
}